// FCG_52115133170149
// MI455X (gfx1250) — compile-verified
//
#include <hip/hip_runtime.h>
#include <hip/hip_bf16.h>

#ifndef __has_builtin
#define __has_builtin(x) 0
#endif

#if defined(__AMDGCN__) && __has_builtin(__builtin_amdgcn_global_load_async_to_lds_b128) && __has_builtin(__builtin_amdgcn_s_wait_asynccnt)
#define HAVE_ASYNC_LDS 1
#else
#define HAVE_ASYNC_LDS 0
#endif

typedef int   v4i __attribute__((ext_vector_type(4)));
typedef float v4f __attribute__((ext_vector_type(4)));

#if HAVE_ASYNC_LDS
typedef __attribute__((address_space(1))) v4i* global_v4i_ptr;
typedef __attribute__((address_space(3))) v4i* lds_v4i_ptr;

__device__ __forceinline__ void async_copy16(const float* g, float* l) {
  // GLOBAL_LOAD_ASYNC_TO_LDS_B128: per-lane 16B global -> LDS, tracked by ASYNCcnt
  __builtin_amdgcn_global_load_async_to_lds_b128(
      (global_v4i_ptr)g, (lds_v4i_ptr)l,
      /*offset=*/0, /*cpol=*/0);
}
#endif

// ---------------- Kernel A: per-parent popcount + block exclusive scan ----------------
// 256 threads, 4 parents/thread -> 1024 parents per block.
// Interior threads do ONE global_load_b128 (4 contiguous occ words) instead of
// four guarded scalar loads (which each forced an s_wait_loadcnt 0).
__global__ void FCG_scanA(const int* __restrict__ xO, int* __restrict__ wsParent,
                          int* __restrict__ wsBlock, int n) {
  __shared__ int sm[256];
  const int tid  = threadIdx.x;
  const int base = blockIdx.x * 1024 + tid * 4;

  int oc[4];
  if (base + 3 < n) {
    v4i o = *(const v4i*)(xO + base);          // single b128 load
    oc[0] = __popc(o.x & 0xFF);
    oc[1] = __popc(o.y & 0xFF);
    oc[2] = __popc(o.z & 0xFF);
    oc[3] = __popc(o.w & 0xFF);
  } else {
#pragma unroll
    for (int i = 0; i < 4; ++i)
      oc[i] = (base + i < n) ? __popc(xO[base + i] & 0xFF) : 0;
  }

  int pre[4];
  int s = 0;
#pragma unroll
  for (int i = 0; i < 4; ++i) { pre[i] = s; s += oc[i]; }

  sm[tid] = s;
  __syncthreads();
  // Hillis-Steele inclusive scan over 256 thread sums
  for (int off = 1; off < 256; off <<= 1) {
    int t = (tid >= off) ? sm[tid - off] : 0;
    __syncthreads();
    sm[tid] += t;
    __syncthreads();
  }
  const int incl = sm[tid];
  const int excl = incl - s;
#pragma unroll
  for (int i = 0; i < 4; ++i) {
    int p = base + i;
    if (p < n) wsParent[p] = excl + pre[i];    // block-local scan; block offset added in scatter
  }
  if (tid == 255) wsBlock[blockIdx.x] = incl;  // block total
}

// ---------------- Kernel B: single-block scan of block totals (nb <= 512) ----------------
__global__ void FCG_scanB(int* __restrict__ wsBlock, int* __restrict__ wsTotal, int nb) {
  __shared__ int sm[256];
  const int tid = threadIdx.x;
  const int i0 = 2 * tid, i1 = 2 * tid + 1;
  const int v0 = (i0 < nb) ? wsBlock[i0] : 0;
  const int v1 = (i1 < nb) ? wsBlock[i1] : 0;
  const int s = v0 + v1;
  sm[tid] = s;
  __syncthreads();
  for (int off = 1; off < 256; off <<= 1) {
    int t = (tid >= off) ? sm[tid - off] : 0;
    __syncthreads();
    sm[tid] += t;
    __syncthreads();
  }
  const int incl = sm[tid];
  const int excl = incl - s;
  if (i0 < nb) wsBlock[i0] = excl;
  if (i1 < nb) wsBlock[i1] = excl + v0;
  if (tid == 255) *wsTotal = incl;             // total kept children K
}

// ---------------- Kernel C: expand + compacted scatter ----------------
// 256 threads = 8 wave32; one wave per parent; lane == feature channel (C == 32 == wave32).
// Feature rows staged global->LDS via async DMA (ASYNCcnt); output stores are
// non-temporal (TH_NT) -- 576MB streamed once, keep it out of L2.
__global__ void FCG_scatter(const int* __restrict__ xC, const int* __restrict__ xO,
                            const float* __restrict__ xF,
                            const int* __restrict__ wsParent, const int* __restrict__ wsBlock,
                            float* __restrict__ outC, float* __restrict__ outF, int n) {
  __shared__ float sF[8 * 32];
  const int tid   = threadIdx.x;
  const int pbase = blockIdx.x * 8;

#if HAVE_ASYNC_LDS
  if (tid < 64) {                              // 64 lanes x 16B = 8 rows x 128B
    int p = pbase + (tid >> 3);
    if (p < n) {
      const float* src = xF + (size_t)pbase * 32 + (size_t)tid * 4;
      async_copy16(src, &sF[tid * 4]);
    }
  }
  __builtin_amdgcn_s_wait_asynccnt(0);
#else
  if (tid < 64) {
    int p = pbase + (tid >> 3);
    if (p < n) {
      const v4f* src = (const v4f*)(xF + (size_t)pbase * 32) + tid;
      ((v4f*)sF)[tid] = *src;
    }
  }
#endif
  __syncthreads();

  const int w = tid >> 5, lane = tid & 31;
  const int p = pbase + w;
  if (p >= n) return;

  const int   occ   = xO[p] & 0xFF;
  int         dst   = wsParent[p] + wsBlock[p >> 10];   // global stable destination
  const float feat  = sF[w * 32 + lane];
  const int   coord = (lane < 4) ? xC[(size_t)p * 4 + lane] : 0;

#pragma unroll
  for (int c = 0; c < 8; ++c) {
    if ((occ >> c) & 1) {
      const size_t r = (size_t)dst;
      __builtin_nontemporal_store(feat, &outF[r * 32 + lane]);  // coalesced 128B row / wave
      if (lane < 4) {
        // child = [batch, 2x + (c&1), 2y + ((c>>1)&1), 2z + ((c>>2)&1)]
        int v = (lane == 0) ? coord : (2 * coord + ((c >> (lane - 1)) & 1));
        __builtin_nontemporal_store((float)v, &outC[r * 4 + lane]);
      }
      ++dst;
    }
  }
}

// ---------------- Kernel D: zero the padding tail (rows >= K), float4 NT stores ----------
// Both regions are 16B-row-aligned: coord row = 1 quad, feature row = 8 quads.
__global__ void FCG_zeroTail(v4f* __restrict__ out4, const int* __restrict__ wsTotal, int n) {
  const long idx  = (long)blockIdx.x * 256 + threadIdx.x;
  const long rows = (long)8 * n;
  const long cQ   = rows;           // coord quads
  const long fQ   = rows * 8;       // feature quads
  if (idx >= cQ + fQ) return;
  const long K = (long)(*wsTotal);
  const v4f z = {0.f, 0.f, 0.f, 0.f};
  if (idx < cQ) {
    if (idx >= K) __builtin_nontemporal_store(z, &out4[idx]);
  } else {
    const long j = idx - cQ;
    if ((j >> 3) >= K) __builtin_nontemporal_store(z, &out4[cQ + j]);
  }
}

extern "C" void kernel_launch(void* const* d_in, const int* in_sizes, int n_in,
                              void* d_out, int out_size, void* d_ws, size_t ws_size,
                              hipStream_t stream) {
  const int*   xC = (const int*)d_in[0];     // [N,4] int32
  const int*   xO = (const int*)d_in[1];     // [N,1] int32
  const float* xF = (const float*)d_in[2];   // [N,C] float32
  const int n = in_sizes[0] / 4;             // N

  float* out  = (float*)d_out;
  float* outC = out;                                   // [8N,4]
  float* outF = out + (size_t)8 * n * 4;               // [8N,32]

  // ws layout (int32): [0..n) parent excl prefixes | [n] total K | [n+1..n+nbA] block sums
  int* ws       = (int*)d_ws;
  int* wsParent = ws;
  int* wsTotal  = ws + n;
  int* wsBlock  = ws + n + 1;

  const int nbA = (n + 1023) / 1024;         // 489 for N=500k (must be <= 512 for scanB)
  FCG_scanA<<<nbA, 256, 0, stream>>>(xO, wsParent, wsBlock, n);
  FCG_scanB<<<1, 256, 0, stream>>>(wsBlock, wsTotal, nbA);

  const int nbC = (n + 7) / 8;
  FCG_scatter<<<nbC, 256, 0, stream>>>(xC, xO, xF, wsParent, wsBlock, outC, outF, n);

  const long totalQuads = (long)8 * n * 9;   // 1 coord quad + 8 feature quads per row
  const int  nbD = (int)((totalQuads + 255) / 256);
  FCG_zeroTail<<<nbD, 256, 0, stream>>>((v4f*)out, wsTotal, n);
}